// multiViewGCN_57741540327593
// MI455X (gfx1250) — compile-verified
//
#include <hip/hip_runtime.h>
#include <math.h>

typedef __attribute__((ext_vector_type(2))) float v2f;
typedef __attribute__((ext_vector_type(8))) float v8f;
typedef __attribute__((ext_vector_type(4))) unsigned int v4u;
typedef __attribute__((ext_vector_type(8))) unsigned int v8u;

#define BATCH 32
#define NNODE 1024
#define NEDGE 8192
#define FIN0  64
#define CDIM  128
#define NPAIR 96
#define KP1   820
#define KP2   656
#define KP3   525

// ---------------------------------------------------------------------------
// f32 WMMA: D(16x16) = A(16x4) * B(4x16) + C   (CDNA5 V_WMMA_F32_16X16X4_F32)
// A layout (ISA 7.12.2): lanes 0-15 -> K=0(v0),K=1(v1); lanes 16-31 -> K=2,K=3
// B/C/D: row striped across lanes within a VGPR; lanes 16-31 carry row+ offset.
// ---------------------------------------------------------------------------
__device__ __forceinline__ v8f wmma4(v2f a, v2f b, v8f c) {
  return __builtin_amdgcn_wmma_f32_16x16x4_f32(
      /*neg_a=*/false, a, /*neg_b=*/false, b,
      /*c_mod=*/(short)0, c, /*reuse_a=*/false, /*reuse_b=*/false);
}

// ---------------------------------------------------------------------------
// TDM 2D tile load: Global -> LDS (CDNA5 ISA).
// D# group0: [count=1 | lds_addr | global_addr(57b) | type=2]
// D# group1: data_size=4B(code 2), tensor_dim0/dim1 (OOB rows read as zero),
//            tile_dim0/tile_dim1, tensor_dim0_stride (elements).
// groups 2/3 zero (2D tensor). Issued wave-uniform; tracked by TENSORcnt.
// ---------------------------------------------------------------------------
__device__ __forceinline__ void tdm_load_2d(unsigned ldsByteAddr, const void* gptr,
                                            unsigned dim0, unsigned dim1,
                                            unsigned tile0, unsigned tile1,
                                            unsigned stride0) {
  unsigned long long ga = (unsigned long long)gptr;
  v4u g0;
  g0[0] = 1u;                                                   // count=1
  g0[1] = ldsByteAddr;                                          // lds_addr
  g0[2] = (unsigned)(ga & 0xffffffffu);                         // global_addr[31:0]
  g0[3] = (unsigned)((ga >> 32) & 0x01ffffffu) | (2u << 30);    // addr[56:32]|type=2
  v8u g1;
  g1[0] = (2u << 16);                                           // data_size = 4B
  g1[1] = (dim0 & 0xffffu) << 16;                               // tensor_dim0 lo
  g1[2] = ((dim0 >> 16) & 0xffffu) | ((dim1 & 0xffffu) << 16);  // dim0 hi | dim1 lo
  g1[3] = ((dim1 >> 16) & 0xffffu) | ((tile0 & 0xffffu) << 16); // dim1 hi | tile0
  g1[4] = (tile1 & 0xffffu);                                    // tile1 | tile2=0
  g1[5] = stride0;                                              // dim0_stride lo32
  g1[6] = 0u;                                                   // stride hi | d1s lo
  g1[7] = 0u;
  v4u gz = {0u, 0u, 0u, 0u};
  asm volatile("tensor_load_to_lds %0, %1, %2, %3"
               :
               : "s"(g0), "s"(g1), "s"(gz), "s"(gz)
               : "memory");
}

// ---------------------------------------------------------------------------
// Stage 0 input packing: x{0,1,2}[B,N,64] -> bufA[p,n,0:64] (pitch CDIM)
// ---------------------------------------------------------------------------
__global__ void pack_input(const float* __restrict__ x0, const float* __restrict__ x1,
                           const float* __restrict__ x2, float* __restrict__ bufA) {
  long t = (long)blockIdx.x * blockDim.x + threadIdx.x;
  const long total = (long)NPAIR * NNODE * FIN0;
  if (t >= total) return;
  int f = (int)(t % FIN0);
  int n = (int)((t / FIN0) % NNODE);
  int p = (int)(t / ((long)FIN0 * NNODE));
  int v = p >> 5, b = p & 31;
  const float* x = (v == 0) ? x0 : (v == 1) ? x1 : x2;
  bufA[((long)p * NNODE + n) * CDIM + f] = x[((long)b * NNODE + n) * FIN0 + f];
}

// ---------------------------------------------------------------------------
// Edge working copies: ei[v][b,2,E] -> src/dst [p,E], mask = 1
// ---------------------------------------------------------------------------
__global__ void init_edges(const int* __restrict__ ei0, const int* __restrict__ ei1,
                           const int* __restrict__ ei2, int* __restrict__ src,
                           int* __restrict__ dst, float* __restrict__ mask) {
  long t = (long)blockIdx.x * blockDim.x + threadIdx.x;
  const long total = (long)NPAIR * NEDGE;
  if (t >= total) return;
  int e = (int)(t % NEDGE);
  int p = (int)(t / NEDGE);
  int v = p >> 5, b = p & 31;
  const int* ei = (v == 0) ? ei0 : (v == 1) ? ei1 : ei2;
  src[t] = ei[(long)b * 2 * NEDGE + e];
  dst[t] = ei[(long)b * 2 * NEDGE + NEDGE + e];
  mask[t] = 1.0f;
}

// ---------------------------------------------------------------------------
// agg[d,:] += x[s,:] * mask   (f32 global atomics; one block per edge)
// ---------------------------------------------------------------------------
__global__ void scatter_add(const float* __restrict__ x, const int* __restrict__ src,
                            const int* __restrict__ dst, const float* __restrict__ mask,
                            float* __restrict__ agg, int Fin) {
  int blk = blockIdx.x;         // p*NEDGE + e
  int f = threadIdx.x;
  if (f >= Fin) return;
  int e = blk % NEDGE;
  int p = blk / NEDGE;
  long eo = (long)p * NEDGE + e;
  float m = mask[eo];
  if (m == 0.0f) return;
  int s = src[eo], d = dst[eo];
  float val = x[((long)p * NNODE + s) * CDIM + f] * m;
  atomicAdd(&agg[((long)p * NNODE + d) * CDIM + f], val);
}

// ---------------------------------------------------------------------------
// Fused GraphConv: out = relu(agg @ Wrel + x @ Wroot + b).
// One 256-thread block (8 waves) per 16-row tile; wave w owns column tile w.
// A-operand slabs (16 x Fin, row pitch CDIM) are DMA'd into LDS by the TDM;
// tensor_dim1 = Nact - row0 makes the TDM zero-fill pad rows (OOB reads = 0).
// ---------------------------------------------------------------------------
__global__ void __launch_bounds__(256)
conv_wmma(const float* __restrict__ agg, const float* __restrict__ xin,
          const float* __restrict__ Wrel, const float* __restrict__ Wroot,
          const float* __restrict__ bias, float* __restrict__ out,
          int Nact, int Fin, int rowTiles) {
  __shared__ float sAgg[16 * CDIM];
  __shared__ float sX[16 * CDIM];
  int tile = blockIdx.x;
  int rt = tile % rowTiles;
  int p  = tile / rowTiles;
  int br = p >> 5;                          // branch
  int row0 = rt * 16;
  unsigned rowsAvail = (unsigned)(Nact - row0);   // >= 1 by construction

  if (threadIdx.x < 32) {                   // wave 0 issues the two TDM DMAs
    const float* aSlab = agg + ((long)p * NNODE + row0) * CDIM;
    const float* xSlab = xin + ((long)p * NNODE + row0) * CDIM;
    tdm_load_2d((unsigned)(unsigned long long)&sAgg[0], aSlab,
                (unsigned)Fin, rowsAvail, (unsigned)Fin, 16u, (unsigned)CDIM);
    tdm_load_2d((unsigned)(unsigned long long)&sX[0], xSlab,
                (unsigned)Fin, rowsAvail, (unsigned)Fin, 16u, (unsigned)CDIM);
    __builtin_amdgcn_s_wait_tensorcnt(0);
  }
  __syncthreads();

  int ct   = threadIdx.x >> 5;              // column tile = wave id (0..7)
  int lane = threadIdx.x & 31;
  int half = lane >> 4;
  int m16  = lane & 15;
  const float* Wr = Wrel  + (long)br * Fin * CDIM + ct * 16 + m16;
  const float* Wo = Wroot + (long)br * Fin * CDIM + ct * 16 + m16;
  v8f acc = {0.f, 0.f, 0.f, 0.f, 0.f, 0.f, 0.f, 0.f};
  for (int k = 0; k < Fin; k += 4) {
    int ka = k + (half << 1);
    v2f a1, a2, b1, b2;
    a1.x = sAgg[m16 * Fin + ka];       a1.y = sAgg[m16 * Fin + ka + 1];
    a2.x = sX[m16 * Fin + ka];         a2.y = sX[m16 * Fin + ka + 1];
    b1.x = Wr[(long)ka * CDIM];        b1.y = Wr[(long)(ka + 1) * CDIM];
    b2.x = Wo[(long)ka * CDIM];        b2.y = Wo[(long)(ka + 1) * CDIM];
    acc = wmma4(a1, b1, acc);
    acc = wmma4(a2, b2, acc);
  }
  int col = ct * 16 + m16;
  float bv = bias[br * CDIM + col];
#pragma unroll
  for (int i = 0; i < 8; ++i) {
    int r = row0 + i + half * 8;            // D: VGPR i -> M=i / M=i+8
    if (r < Nact) {
      float vv = acc[i] + bv;
      out[((long)p * NNODE + r) * CDIM + col] = vv > 0.0f ? vv : 0.0f;
    }
  }
}

// ---------------------------------------------------------------------------
// TopKPooling: score = tanh(x.w/||w||); full in-LDS bitonic sort of 1024
// (score, idx) pairs (desc, idx tiebreak asc); emit perm, kept scores, inv map.
// ---------------------------------------------------------------------------
__global__ void __launch_bounds__(256)
topk_sort(const float* __restrict__ xin, const float* __restrict__ poolw,
          int Nact, int K, float* __restrict__ scoreKept,
          int* __restrict__ perm, int* __restrict__ inv) {
  __shared__ float skey[NNODE];
  __shared__ int   sidx[NNODE];
  __shared__ float wsh[CDIM];
  __shared__ float red[CDIM];
  int p = blockIdx.x;
  int br = p >> 5;
  int tid = threadIdx.x;
  if (tid < CDIM) {
    float w = poolw[br * CDIM + tid];
    wsh[tid] = w;
    red[tid] = w * w;
  }
  __syncthreads();
  for (int s = CDIM / 2; s > 0; s >>= 1) {
    if (tid < s) red[tid] += red[tid + s];
    __syncthreads();
  }
  float invn = rsqrtf(red[0]);
  for (int n = tid; n < NNODE; n += 256) {
    float sc = -1e30f;
    if (n < Nact) {
      const float* xr = xin + ((long)p * NNODE + n) * CDIM;
      float d = 0.0f;
      for (int c = 0; c < CDIM; ++c) d += xr[c] * wsh[c];
      sc = tanhf(d * invn);
    }
    skey[n] = sc;
    sidx[n] = n;
  }
  __syncthreads();
  for (int k = 2; k <= NNODE; k <<= 1) {
    for (int j = k >> 1; j > 0; j >>= 1) {
      for (int i = tid; i < NNODE; i += 256) {
        int ixj = i ^ j;
        if (ixj > i) {
          float a = skey[i], b = skey[ixj];
          int ia = sidx[i], ib = sidx[ixj];
          bool aFirst = (a > b) || (a == b && ia < ib);
          bool desc = ((i & k) == 0);
          if (desc ? !aFirst : aFirst) {
            skey[i] = b; skey[ixj] = a;
            sidx[i] = ib; sidx[ixj] = ia;
          }
        }
      }
      __syncthreads();
    }
  }
  for (int i = tid; i < NNODE; i += 256) inv[(long)p * NNODE + i] = -1;
  __syncthreads();
  for (int i = tid; i < NNODE; i += 256) {
    scoreKept[(long)p * NNODE + i] = skey[i];
    perm[(long)p * NNODE + i] = sidx[i];
    if (i < K) inv[(long)p * NNODE + sidx[i]] = i;
  }
}

// xout[i,:] = xin[perm[i],:] * score[i]
__global__ void gather_scale(const float* __restrict__ xin,
                             const float* __restrict__ scoreKept,
                             const int* __restrict__ perm,
                             float* __restrict__ xout, int K) {
  int blk = blockIdx.x;        // p*K + i
  int c = threadIdx.x;         // CDIM
  int i = blk % K;
  int p = blk / K;
  int pi = perm[(long)p * NNODE + i];
  float sc = scoreKept[(long)p * NNODE + i];
  xout[((long)p * NNODE + i) * CDIM + c] =
      xin[((long)p * NNODE + pi) * CDIM + c] * sc;
}

__global__ void relabel_edges(int* __restrict__ src, int* __restrict__ dst,
                              float* __restrict__ mask, const int* __restrict__ inv) {
  long t = (long)blockIdx.x * blockDim.x + threadIdx.x;
  const long total = (long)NPAIR * NEDGE;
  if (t >= total) return;
  int p = (int)(t / NEDGE);
  int ns = inv[(long)p * NNODE + src[t]];
  int nd = inv[(long)p * NNODE + dst[t]];
  bool valid = (ns >= 0) && (nd >= 0);
  src[t] = valid ? ns : 0;
  dst[t] = valid ? nd : 0;
  if (!valid) mask[t] = 0.0f;
}

// feats[p, 0:128] += max over nodes;  feats[p, 128:256] += mean over nodes
__global__ void readout_acc(const float* __restrict__ x, int K, float* __restrict__ feats) {
  int p = blockIdx.x;
  int c = threadIdx.x;         // CDIM
  const float* xp = x + (long)p * NNODE * CDIM + c;
  float mx = -1e30f, sm = 0.0f;
  for (int i = 0; i < K; ++i) {
    float v = xp[(long)i * CDIM];
    mx = fmaxf(mx, v);
    sm += v;
  }
  feats[(long)p * 2 * CDIM + c] += mx;
  feats[(long)p * 2 * CDIM + CDIM + c] += sm / (float)K;
}

// h0[b, v*256 + j] = feats[v*32+b, j]
__global__ void pack_feats(const float* __restrict__ feats, float* __restrict__ h0) {
  int t = blockIdx.x * blockDim.x + threadIdx.x;
  const int total = BATCH * 768;
  if (t >= total) return;
  int b = t / 768;
  int j = t % 768;
  int v = j / 256;
  int jj = j % 256;
  h0[t] = feats[((long)(v * BATCH + b)) * 256 + jj];
}

// Y[32,Nc] = act(X[32,Kdim] @ W[Kdim,Nc] + bias); one wave per 16x16 tile
__global__ void __launch_bounds__(32)
mlp_gemm_wmma(const float* __restrict__ X, const float* __restrict__ W,
              const float* __restrict__ bias, float* __restrict__ Y,
              int Kdim, int Nc, int doRelu) {
  int blk = blockIdx.x;
  int rt = blk & 1;            // 32 rows -> 2 row tiles
  int ct = blk >> 1;
  int lane = threadIdx.x;
  int half = lane >> 4;
  int m16  = lane & 15;
  int row = rt * 16 + m16;
  const float* Xp = X + (long)row * Kdim;
  const float* Wp = W + ct * 16 + m16;
  v8f acc = {0.f, 0.f, 0.f, 0.f, 0.f, 0.f, 0.f, 0.f};
  for (int k = 0; k < Kdim; k += 4) {
    int ka = k + (half << 1);
    v2f a, b;
    a.x = Xp[ka];               a.y = Xp[ka + 1];
    b.x = Wp[(long)ka * Nc];    b.y = Wp[(long)(ka + 1) * Nc];
    acc = wmma4(a, b, acc);
  }
  int col = ct * 16 + m16;
  float bv = bias[col];
#pragma unroll
  for (int i = 0; i < 8; ++i) {
    int r = rt * 16 + i + half * 8;
    float vv = acc[i] + bv;
    if (doRelu) vv = vv > 0.0f ? vv : 0.0f;
    Y[(long)r * Nc + col] = vv;
  }
}

// out[b,j] = (h2[b,:] . W2[:,j] + b2[j]) / 1000
__global__ void final_mlp(const float* __restrict__ h2, const float* __restrict__ W2,
                          const float* __restrict__ b2, float* __restrict__ out) {
  int t = threadIdx.x;
  if (t >= BATCH * 5) return;
  int b = t / 5, j = t % 5;
  float acc = b2[j];
  const float* hr = h2 + b * CDIM;
  for (int k = 0; k < CDIM; ++k) acc += hr[k] * W2[k * 5 + j];
  out[t] = acc * 0.001f;
}

// ---------------------------------------------------------------------------
extern "C" void kernel_launch(void* const* d_in, const int* in_sizes, int n_in,
                              void* d_out, int out_size, void* d_ws, size_t ws_size,
                              hipStream_t stream) {
  (void)in_sizes; (void)n_in; (void)out_size; (void)ws_size;
  const float* x0 = (const float*)d_in[0];
  const float* x1 = (const float*)d_in[1];
  const float* x2 = (const float*)d_in[2];
  const int* ei0 = (const int*)d_in[3];
  const int* ei1 = (const int*)d_in[4];
  const int* ei2 = (const int*)d_in[5];
  const float* c0Wrel = (const float*)d_in[6];
  const float* c0Wroot = (const float*)d_in[7];
  const float* c0b = (const float*)d_in[8];
  const float* p0w = (const float*)d_in[9];
  const float* c1Wrel = (const float*)d_in[10];
  const float* c1Wroot = (const float*)d_in[11];
  const float* c1b = (const float*)d_in[12];
  const float* p1w = (const float*)d_in[13];
  const float* c2Wrel = (const float*)d_in[14];
  const float* c2Wroot = (const float*)d_in[15];
  const float* c2b = (const float*)d_in[16];
  const float* p2w = (const float*)d_in[17];
  const float* mlp0W = (const float*)d_in[18];
  const float* mlp0b = (const float*)d_in[19];
  const float* mlp1W = (const float*)d_in[20];
  const float* mlp1b = (const float*)d_in[21];
  const float* mlp2W = (const float*)d_in[22];
  const float* mlp2b = (const float*)d_in[23];
  float* out = (float*)d_out;

  char* ws = (char*)d_ws;
  size_t off = 0;
  auto wsalloc = [&](size_t bytes) -> void* {
    void* p = ws + off;
    off += (bytes + 255) & ~(size_t)255;
    return p;
  };
  const size_t nodeBytes = (size_t)NPAIR * NNODE * CDIM * sizeof(float);
  float* bufA   = (float*)wsalloc(nodeBytes);
  float* bufB   = (float*)wsalloc(nodeBytes);
  float* aggBuf = (float*)wsalloc(nodeBytes);
  int*   srcw   = (int*)  wsalloc((size_t)NPAIR * NEDGE * sizeof(int));
  int*   dstw   = (int*)  wsalloc((size_t)NPAIR * NEDGE * sizeof(int));
  float* maskw  = (float*)wsalloc((size_t)NPAIR * NEDGE * sizeof(float));
  float* scorew = (float*)wsalloc((size_t)NPAIR * NNODE * sizeof(float));
  int*   permw  = (int*)  wsalloc((size_t)NPAIR * NNODE * sizeof(int));
  int*   invw   = (int*)  wsalloc((size_t)NPAIR * NNODE * sizeof(int));
  float* feats  = (float*)wsalloc((size_t)NPAIR * 2 * CDIM * sizeof(float));
  float* h0     = (float*)wsalloc((size_t)BATCH * 768 * sizeof(float));
  float* h1     = (float*)wsalloc((size_t)BATCH * 512 * sizeof(float));
  float* h2     = (float*)wsalloc((size_t)BATCH * CDIM * sizeof(float));

  const long packTotal = (long)NPAIR * NNODE * FIN0;
  const long edgeTotal = (long)NPAIR * NEDGE;

  pack_input<<<(unsigned)((packTotal + 255) / 256), 256, 0, stream>>>(x0, x1, x2, bufA);
  init_edges<<<(unsigned)((edgeTotal + 255) / 256), 256, 0, stream>>>(ei0, ei1, ei2,
                                                                      srcw, dstw, maskw);
  hipMemsetAsync(feats, 0, (size_t)NPAIR * 2 * CDIM * sizeof(float), stream);

  struct Stage {
    const float *Wrel, *Wroot, *bias, *pw;
    int Nact, Fin, K;
  } stages[3] = {
      {c0Wrel, c0Wroot, c0b, p0w, NNODE, FIN0, KP1},
      {c1Wrel, c1Wroot, c1b, p1w, KP1,   CDIM, KP2},
      {c2Wrel, c2Wroot, c2b, p2w, KP2,   CDIM, KP3},
  };

  for (int s = 0; s < 3; ++s) {
    const Stage& st = stages[s];
    int rowTiles = (st.Nact + 15) / 16;
    hipMemsetAsync(aggBuf, 0, nodeBytes, stream);
    scatter_add<<<(unsigned)edgeTotal, st.Fin, 0, stream>>>(bufA, srcw, dstw, maskw,
                                                            aggBuf, st.Fin);
    conv_wmma<<<(unsigned)(NPAIR * rowTiles), 256, 0, stream>>>(
        aggBuf, bufA, st.Wrel, st.Wroot, st.bias, bufB, st.Nact, st.Fin, rowTiles);
    topk_sort<<<NPAIR, 256, 0, stream>>>(bufB, st.pw, st.Nact, st.K, scorew, permw, invw);
    gather_scale<<<(unsigned)(NPAIR * st.K), CDIM, 0, stream>>>(bufB, scorew, permw,
                                                                bufA, st.K);
    if (s < 2)
      relabel_edges<<<(unsigned)((edgeTotal + 255) / 256), 256, 0, stream>>>(srcw, dstw,
                                                                             maskw, invw);
    readout_acc<<<NPAIR, CDIM, 0, stream>>>(bufA, st.K, feats);
  }

  pack_feats<<<(BATCH * 768 + 255) / 256, 256, 0, stream>>>(feats, h0);
  mlp_gemm_wmma<<<2 * (512 / 16), 32, 0, stream>>>(h0, mlp0W, mlp0b, h1, 768, 512, 1);
  mlp_gemm_wmma<<<2 * (CDIM / 16), 32, 0, stream>>>(h1, mlp1W, mlp1b, h2, 512, CDIM, 1);
  final_mlp<<<1, 256, 0, stream>>>(h2, mlp2W, mlp2b, out);
}